// Router_6768868459217
// MI455X (gfx1250) — compile-verified
//
#include <hip/hip_runtime.h>
#include <math.h>

typedef float v2f __attribute__((ext_vector_type(2)));
typedef float v4f __attribute__((ext_vector_type(4)));
typedef float v8f __attribute__((ext_vector_type(8)));

#define T_TOKENS 32768
#define D_DIM    1024
#define E_EXP    16
#define TOPK     2
#define TK       (T_TOKENS * TOPK)   // 65536 slots
#define NSTEPS   (D_DIM / 4)         // 256 WMMA k-steps
#define ROWS_PER_BLOCK 4

// ---------------------------------------------------------------------------
// Kernel 0: pre-pack router weights into the exact WMMA B-operand lane layout.
// B (4x16, K x N): VGPR0 = rows K=k0 (lanes0-15) / K=k0+1 (lanes16-31),
//                  VGPR1 = rows K=k0+2 / k0+3 ; B[k][n] = W[n][k].
// Per (step, lane): b.x = W[e][k0+khalf], b.y = W[e][k0+khalf+2].
// 64KB table -> resident in L2/WGP$, read as contiguous per-lane b64.
// ---------------------------------------------------------------------------
__global__ __launch_bounds__(256) void prepack_kernel(
    const float* __restrict__ W, float* __restrict__ Bp)
{
    const int idx   = blockIdx.x * 256 + threadIdx.x;  // 0 .. 8191
    const int step  = idx >> 5;
    const int lane  = idx & 31;
    const int e     = lane & 15;
    const int khalf = lane >> 4;
    const int k0    = step * 4;
    Bp[idx * 2 + 0] = W[e * D_DIM + k0 + khalf];
    Bp[idx * 2 + 1] = W[e * D_DIM + k0 + khalf + 2];
}

// ---------------------------------------------------------------------------
// Kernel 1: router logits via V_WMMA_F32_16X16X4_F32, sigmoid, biased top-2.
// One wave handles a 16-token tile; 8 waves (256 thr) per block.
// Inner loop batch-unrolled x8: 16 clustered loads, then 8 chained WMMAs.
// ---------------------------------------------------------------------------
__global__ __launch_bounds__(256) void router_kernel(
    const float* __restrict__ x,          // [T, D]
    const float* __restrict__ Bp,         // [NSTEPS*32] v2f pre-packed B
    const float* __restrict__ bias,       // [E]
    int*   __restrict__ sel,              // [T*2] selected expert per slot
    float* __restrict__ scoresFlat)       // [T*2] normalized score per slot
{
    __shared__ float lds[8][16 * 16];     // per-wave 16x16 logit tile

    const int wave   = threadIdx.x >> 5;
    const int lane   = threadIdx.x & 31;
    const int row16  = lane & 15;
    const int khalf  = lane >> 4;
    const int tile   = blockIdx.x * 8 + wave;
    const int token0 = tile * 16;

    const float* xrow = x + (size_t)(token0 + row16) * D_DIM + khalf * 2;
    const v2f*   bp   = (const v2f*)Bp + lane;          // + step*32 per step

    v8f c = {0.f, 0.f, 0.f, 0.f, 0.f, 0.f, 0.f, 0.f};

    for (int k0 = 0; k0 < D_DIM; k0 += 32) {            // 8 steps per batch
        v2f a[8], b[8];
#pragma unroll
        for (int u = 0; u < 8; ++u) {
            a[u] = *(const v2f*)(xrow + k0 + u * 4);    // A: 2 contiguous K
            b[u] = bp[((k0 >> 2) + u) * 32];            // B: pre-packed lane
        }
#pragma unroll
        for (int u = 0; u < 8; ++u)
            c = __builtin_amdgcn_wmma_f32_16x16x4_f32(
                    false, a[u], false, b[u], (short)0, c, false, false);
    }

    // C/D layout: VGPR r -> M = r (lanes 0-15) or r+8 (lanes 16-31), N = row16
#pragma unroll
    for (int r = 0; r < 8; ++r)
        lds[wave][(r + khalf * 8) * 16 + row16] = c[r];
    __syncthreads();

    if (lane < 16) {
        const int t = token0 + row16;     // this thread's token
        float s[E_EXP];
#pragma unroll
        for (int e = 0; e < E_EXP; ++e) {
            float l = lds[wave][row16 * 16 + e];
            s[e] = 1.0f / (1.0f + expf(-l));
        }
        // biased top-1 (ties -> lowest index, matches jax.lax.top_k)
        int i1 = 0; float b1 = s[0] + bias[0];
#pragma unroll
        for (int e = 1; e < E_EXP; ++e) {
            float bv = s[e] + bias[e];
            if (bv > b1) { b1 = bv; i1 = e; }
        }
        // biased top-2 excluding i1
        int i2 = -1; float b2 = -INFINITY;
#pragma unroll
        for (int e = 0; e < E_EXP; ++e) {
            if (e == i1) continue;
            float bv = s[e] + bias[e];
            if (bv > b2) { b2 = bv; i2 = e; }
        }
        const float s1 = s[i1], s2 = s[i2];
        const float inv = 1.0f / (s1 + s2);
        sel[2 * t + 0]        = i1;
        sel[2 * t + 1]        = i2;
        scoresFlat[2 * t + 0] = s1 * inv;
        scoresFlat[2 * t + 1] = s2 * inv;
    }
}

// ---------------------------------------------------------------------------
// Kernel 2: per-block expert histograms (256 slots per block, 256 blocks)
// ---------------------------------------------------------------------------
__global__ __launch_bounds__(256) void hist_kernel(
    const int* __restrict__ sel, int* __restrict__ blockHist)
{
    __shared__ int h[E_EXP];
    if (threadIdx.x < E_EXP) h[threadIdx.x] = 0;
    __syncthreads();
    const int i = blockIdx.x * 256 + threadIdx.x;
    atomicAdd(&h[sel[i]], 1);
    __syncthreads();
    if (threadIdx.x < E_EXP)
        blockHist[blockIdx.x * E_EXP + threadIdx.x] = h[threadIdx.x];
}

// ---------------------------------------------------------------------------
// Kernel 3: expert totals + exclusive scan over experts and blocks.
// blockHist is rewritten in place with each block's global start per expert.
// ---------------------------------------------------------------------------
__global__ __launch_bounds__(64) void scan_kernel(
    int* __restrict__ blockHist, float* __restrict__ outCounts)
{
    __shared__ int totals[E_EXP];
    __shared__ int base[E_EXP];
    const int e = threadIdx.x;
    if (e < E_EXP) {
        int t = 0;
        for (int b = 0; b < 256; ++b) t += blockHist[b * E_EXP + e];
        totals[e] = t;
        outCounts[e] = (float)t;          // num_tokens_per_expert (as float)
    }
    __syncthreads();
    if (threadIdx.x == 0) {
        int acc = 0;
        for (int k = 0; k < E_EXP; ++k) { base[k] = acc; acc += totals[k]; }
    }
    __syncthreads();
    if (e < E_EXP) {
        int run = base[e];
        for (int b = 0; b < 256; ++b) {
            int v = blockHist[b * E_EXP + e];
            blockHist[b * E_EXP + e] = run;
            run += v;
        }
    }
}

// ---------------------------------------------------------------------------
// Kernel 4: stable counting-sort scatter. dst is a bijection over [0, TK).
// ---------------------------------------------------------------------------
__global__ __launch_bounds__(256) void scatter_kernel(
    const int*   __restrict__ sel,
    const float* __restrict__ scoresFlat,
    const int*   __restrict__ blockStart,
    int*   __restrict__ perm,
    float* __restrict__ outScatter,
    float* __restrict__ outScores)
{
    __shared__ int se[256];
    const int i = blockIdx.x * 256 + threadIdx.x;
    const int e = sel[i];
    se[threadIdx.x] = e;
    __syncthreads();
    int rank = 0;
    for (int j = 0; j < (int)threadIdx.x; ++j) rank += (se[j] == e);
    const int dst = blockStart[blockIdx.x * E_EXP + e] + rank;
    perm[dst]       = i;
    outScatter[dst] = (float)(i >> 1);    // scatter_indices = slot // top_k
    outScores[dst]  = scoresFlat[i];
}

// ---------------------------------------------------------------------------
// Kernel 5: gather rows (dominant cost, ~384MB traffic). x hits L2 (128MB
// < 192MB); output streamed with non-temporal b128 stores so it does not
// evict x. 4 rows per block -> 4 independent loads/stores per thread.
// ---------------------------------------------------------------------------
__global__ __launch_bounds__(256) void gather_kernel(
    const float* __restrict__ x, const int* __restrict__ perm,
    float* __restrict__ outX)
{
    const int p0 = blockIdx.x * ROWS_PER_BLOCK;
    int tok[ROWS_PER_BLOCK];
#pragma unroll
    for (int u = 0; u < ROWS_PER_BLOCK; ++u)
        tok[u] = perm[p0 + u] >> 1;
    v4f v[ROWS_PER_BLOCK];
#pragma unroll
    for (int u = 0; u < ROWS_PER_BLOCK; ++u)
        v[u] = ((const v4f*)(x + (size_t)tok[u] * D_DIM))[threadIdx.x];
#pragma unroll
    for (int u = 0; u < ROWS_PER_BLOCK; ++u)
        __builtin_nontemporal_store(
            v[u], &((v4f*)(outX + (size_t)(p0 + u) * D_DIM))[threadIdx.x]);
}

// ---------------------------------------------------------------------------
extern "C" void kernel_launch(void* const* d_in, const int* in_sizes, int n_in,
                              void* d_out, int out_size, void* d_ws, size_t ws_size,
                              hipStream_t stream) {
    const float* x    = (const float*)d_in[0];   // [32768, 1024]
    const float* W    = (const float*)d_in[1];   // [16, 1024]
    const float* bias = (const float*)d_in[2];   // [16]
    // d_in[3] = top_k scalar (fixed = 2)

    float* out        = (float*)d_out;
    float* outX       = out;                               // 65536*1024
    float* outCounts  = out + (size_t)TK * D_DIM;          // 16
    float* outScatter = outCounts + E_EXP;                 // 65536
    float* outScores  = outScatter + TK;                   // 65536

    int*   sel        = (int*)d_ws;                        // TK ints
    int*   perm       = sel + TK;                          // TK ints
    int*   blockHist  = perm + TK;                         // 256*16 ints
    float* scoresFlat = (float*)(blockHist + 256 * E_EXP); // TK floats
    float* Bpacked    = scoresFlat + TK;                   // NSTEPS*64 floats

    prepack_kernel<<<NSTEPS * 32 / 256,   256, 0, stream>>>(W, Bpacked);
    router_kernel <<<T_TOKENS / (16 * 8), 256, 0, stream>>>(x, Bpacked, bias, sel, scoresFlat);
    hist_kernel   <<<TK / 256,            256, 0, stream>>>(sel, blockHist);
    scan_kernel   <<<1,                    64, 0, stream>>>(blockHist, outCounts);
    scatter_kernel<<<TK / 256,            256, 0, stream>>>(sel, scoresFlat, blockHist,
                                                            perm, outScatter, outScores);
    gather_kernel <<<TK / ROWS_PER_BLOCK, 256, 0, stream>>>(x, perm, outX);
}